// User2ItemNet_661424964276
// MI455X (gfx1250) — compile-verified
//
#include <hip/hip_runtime.h>
#include <hip/hip_bf16.h>

typedef __attribute__((ext_vector_type(2))) float v2f;
typedef __attribute__((ext_vector_type(4))) float v4f;
typedef __attribute__((ext_vector_type(8))) float v8f;

#define WMMA4(A, Bv, C) \
  __builtin_amdgcn_wmma_f32_16x16x4_f32(false, (A), false, (Bv), (short)0, (C), false, false)

// Problem constants
#define BB   4096
#define LL   200
#define DD   64
#define CINK 68     // CTX(64) + POS(4)
#define AINK 256
#define H1N  80
#define H2N  40

// LDS layout (float offsets)
#define S_OFF   0        // sequence_emb: 208 rows x stride 68
#define S_STR   68
#define WQ_OFF  14144    // wq^T: 64 rows x stride 68 (row n, col k)
#define W1_OFF  18496    // w1^T: 80 rows x stride 260
#define W2_OFF  39296    // w2^T (zero padded to N=48): 48 rows x stride 84
#define SCR_OFF 43328    // per-wave scratch: 8 x 1344 floats (16x68 q / 16x84 h1)
#define SCO_OFF 54080    // scores: 256
#define BQ_OFF  54336
#define AQ_OFF  54400
#define B1_OFF  54464
#define B2_OFF  54544    // padded to 48
#define W3_OFF  54592    // padded to 48
#define BO_OFF  54640
#define AO_OFF  54704
#define B3_OFF  54768
#define U_OFF   54784
#define USR_OFF 54848
#define RED_OFF 54912
#define PR_OFF  55168
#define PP_OFF  55232
#define PN_OFF  55296
#define SMEM_FLOATS 55360
#define SMEM_BYTES  (SMEM_FLOATS * 4)   // 221440 bytes

__device__ __forceinline__ float sigmoidf_(float x) {
  return 1.0f / (1.0f + __expf(-x));
}

__global__ __launch_bounds__(256) void din_main(
    const float* __restrict__ target, const float* __restrict__ seq,
    const float* __restrict__ ctx,    const float* __restrict__ seq2,
    const float* __restrict__ neg,    const int*   __restrict__ mask,
    const float* __restrict__ pos,    const float* __restrict__ wq,
    const float* __restrict__ bq,     const float* __restrict__ aq,
    const float* __restrict__ w1,     const float* __restrict__ b1,
    const float* __restrict__ w2,     const float* __restrict__ b2,
    const float* __restrict__ w3,     const float* __restrict__ b3,
    const float* __restrict__ wo,     const float* __restrict__ bo,
    const float* __restrict__ ao,     float* __restrict__ out,
    float* __restrict__ ws) {
  extern __shared__ float sm[];
  const int b = blockIdx.x;
  const int t = threadIdx.x;

  // ---- Stage per-batch sequence (200x64 -> stride 68, pad rows 200..207 with 0)
  {
    const float* sb = seq + (size_t)b * (LL * DD);
    for (int e4 = t; e4 < (LL * DD) / 4; e4 += 256) {
      int e = e4 * 4;
      int r = e >> 6, c = e & 63;
      *(v4f*)&sm[S_OFF + r * S_STR + c] = *(const v4f*)&sb[e];
    }
    for (int e = t; e < 8 * S_STR; e += 256)
      sm[S_OFF + (200 + e / S_STR) * S_STR + (e % S_STR)] = 0.0f;
  }
  // ---- Stage transposed weights
  for (int e = t; e < CINK * DD; e += 256) {       // wq (68,64) -> wq_t[n][k]
    int k = e >> 6, n = e & 63;
    sm[WQ_OFF + n * 68 + k] = wq[e];
  }
  for (int e = t; e < AINK * H1N; e += 256) {      // w1 (256,80) -> w1_t[n][k]
    int k = e / H1N, n = e % H1N;
    sm[W1_OFF + n * 260 + k] = w1[e];
  }
  for (int e = t; e < 48 * 84; e += 256) {         // w2 (80,40) -> w2_t[n][k], zero pad
    int n = e / 84, k = e % 84;
    sm[W2_OFF + e] = (n < H2N && k < H1N) ? w2[k * H2N + n] : 0.0f;
  }
  // ---- Stage biases / slopes
  if (t < 64) {
    sm[BQ_OFF + t] = bq[t];
    sm[AQ_OFF + t] = aq[t];
    sm[BO_OFF + t] = bo[t];
    sm[AO_OFF + t] = ao[t];
  }
  if (t < 80) sm[B1_OFF + t] = b1[t];
  if (t < 48) {
    sm[B2_OFF + t] = (t < H2N) ? b2[t] : 0.0f;
    sm[W3_OFF + t] = (t < H2N) ? w3[t] : 0.0f;
  }
  if (t == 0) sm[B3_OFF] = b3[0];
  __syncthreads();

  const int wave = t >> 5;
  const int lane = t & 31;
  const int half = lane >> 4;   // 0: lanes 0-15, 1: lanes 16-31
  const int mr   = lane & 15;
  float* scr = &sm[SCR_OFF + wave * 1344];

  // ---- Fused per-M-tile MLP (13 tiles of 16 rows, padded to 208)
  for (int mt = wave; mt < 13; mt += 8) {
    const int m = mt * 16 + mr;
    const float validf = (m < LL) ? 1.0f : 0.0f;   // branch-free row padding
    const int   msafe  = (m < LL) ? m : 0;         // clamped (always-legal) row

    // GEMM1: query = prelu(ctx @ wq + bq), K=68 (17 k-steps), N=64 (4 tiles)
    v8f acc1[4] = {};
    // peeled ks=0: first 4 K-values come from pos_emb (POS=4 aligns with k-step)
    {
      const int k = half * 2;
      v2f a = *(const v2f*)(pos + msafe * 4 + k);
      a *= validf;
#pragma unroll
      for (int nt = 0; nt < 4; ++nt) {
        v2f bv = *(const v2f*)(&sm[WQ_OFF + (nt * 16 + mr) * 68 + k]);
        acc1[nt] = WMMA4(a, bv, acc1[nt]);
      }
    }
    const float* actx = ctx + ((size_t)b * LL + msafe) * 64;
#pragma unroll 1
    for (int ks = 1; ks < 17; ++ks) {
      const int k = ks * 4 + half * 2;
      v2f a = *(const v2f*)(actx + (k - 4));
      a *= validf;
#pragma unroll
      for (int nt = 0; nt < 4; ++nt) {
        v2f bv = *(const v2f*)(&sm[WQ_OFF + (nt * 16 + mr) * 68 + k]);
        acc1[nt] = WMMA4(a, bv, acc1[nt]);
      }
    }
    // bias + prelu -> q scratch (16 x stride 68)
#pragma unroll
    for (int nt = 0; nt < 4; ++nt) {
      const int n = nt * 16 + mr;
      const float bqv = sm[BQ_OFF + n], aqv = sm[AQ_OFF + n];
#pragma unroll
      for (int r = 0; r < 8; ++r) {
        float v = acc1[nt][r] + bqv;
        v = (v >= 0.0f) ? v : aqv * v;
        scr[(r + half * 8) * 68 + n] = v;   // same-wave DS: ordered, no barrier
      }
    }

    // GEMM2: h1 = sigmoid(inp @ w1 + b1), K=256 (64 k-steps), N=80 (5 tiles)
    // inp[k] = k<64: q | 64..127: s | 128..191: q-s | 192..255: q*s
    v8f acc2[5] = {};
#pragma unroll 1
    for (int ks = 0; ks < 64; ++ks) {
      const int k = ks * 4 + half * 2;
      const int region = k >> 6;
      const int kk = k & 63;
      v2f qv = *(const v2f*)(&scr[mr * 68 + kk]);
      v2f sv = *(const v2f*)(&sm[S_OFF + (mt * 16 + mr) * S_STR + kk]);
      v2f a = (region == 0) ? qv
            : (region == 1) ? sv
            : (region == 2) ? (qv - sv)
                            : (qv * sv);
#pragma unroll
      for (int nt = 0; nt < 5; ++nt) {
        v2f bv = *(const v2f*)(&sm[W1_OFF + (nt * 16 + mr) * 260 + k]);
        acc2[nt] = WMMA4(a, bv, acc2[nt]);
      }
    }
    // bias + sigmoid -> h1 scratch (16 x stride 84) [q scratch dead now]
#pragma unroll
    for (int nt = 0; nt < 5; ++nt) {
      const int n = nt * 16 + mr;
      const float b1v = sm[B1_OFF + n];
#pragma unroll
      for (int r = 0; r < 8; ++r) {
        float v = sigmoidf_(acc2[nt][r] + b1v);
        scr[(r + half * 8) * 84 + n] = v;
      }
    }

    // GEMM3: h2 = h1 @ w2, K=80 (20 k-steps), N padded to 48 (3 tiles)
    v8f acc3[3] = {};
#pragma unroll 1
    for (int ks = 0; ks < 20; ++ks) {
      const int k = ks * 4 + half * 2;
      v2f a = *(const v2f*)(&scr[mr * 84 + k]);
#pragma unroll
      for (int nt = 0; nt < 3; ++nt) {
        v2f bv = *(const v2f*)(&sm[W2_OFF + (nt * 16 + mr) * 84 + k]);
        acc3[nt] = WMMA4(a, bv, acc3[nt]);
      }
    }
    // score[m] = sum_n sigmoid(h2+b2) * w3  (pads contribute 0 since w3e=0)
#pragma unroll
    for (int r = 0; r < 8; ++r) {
      float p = 0.0f;
#pragma unroll
      for (int nt = 0; nt < 3; ++nt) {
        const int n = nt * 16 + mr;
        p += sigmoidf_(acc3[nt][r] + sm[B2_OFF + n]) * sm[W3_OFF + n];
      }
      // reduce across the 16 lanes of each half
      p += __shfl_xor(p, 1, 32);
      p += __shfl_xor(p, 2, 32);
      p += __shfl_xor(p, 4, 32);
      p += __shfl_xor(p, 8, 32);
      if (mr == 0)
        sm[SCO_OFF + mt * 16 + r + half * 8] = p + sm[B3_OFF];
    }
  }
  __syncthreads();

  // ---- Softmax over 200 scores (deterministic tree reductions)
  float x = (t < LL) ? sm[SCO_OFF + t] : -3.0e38f;
  sm[RED_OFF + t] = x;
  __syncthreads();
  for (int s2 = 128; s2 > 0; s2 >>= 1) {
    if (t < s2) sm[RED_OFF + t] = fmaxf(sm[RED_OFF + t], sm[RED_OFF + t + s2]);
    __syncthreads();
  }
  const float mx = sm[RED_OFF];
  __syncthreads();
  float e = (t < LL) ? __expf(x - mx) : 0.0f;
  sm[RED_OFF + t] = e;
  __syncthreads();
  for (int s2 = 128; s2 > 0; s2 >>= 1) {
    if (t < s2) sm[RED_OFF + t] += sm[RED_OFF + t + s2];
    __syncthreads();
  }
  const float den = sm[RED_OFF];
  __syncthreads();
  sm[SCO_OFF + t] = e / den;
  __syncthreads();

  // ---- u = sum_l w[l] * s[l][:]  (64 dims)
  if (t < 64) {
    float acc = 0.0f;
    for (int l = 0; l < LL; ++l)
      acc += sm[SCO_OFF + l] * sm[S_OFF + l * S_STR + t];
    sm[U_OFF + t] = acc;
  }
  __syncthreads();

  // ---- user = prelu(u @ wo + bo); partial products for final dots
  if (t < 64) {
    float v = 0.0f;
    for (int k = 0; k < 64; ++k)
      v += sm[U_OFF + k] * wo[k * 64 + t];
    v += sm[BO_OFF + t];
    v = (v >= 0.0f) ? v : sm[AO_OFF + t] * v;
    sm[USR_OFF + t] = v;
    sm[PR_OFF + t] = v * target[(size_t)b * 64 + t];
    sm[PP_OFF + t] = v * seq2[((size_t)b * LL + (LL - 1)) * 64 + t];
    sm[PN_OFF + t] = v * neg[(size_t)b * 64 + t];
  }
  __syncthreads();

  if (t == 0) {
    float sr = 0.0f, sp = 0.0f, sn = 0.0f;
    for (int i = 0; i < 64; ++i) {
      sr += sm[PR_OFF + i];
      sp += sm[PP_OFF + i];
      sn += sm[PN_OFF + i];
    }
    out[b] = sr;  // rel_u2i
    // aux = -log(sigmoid(sp)) - log(1-sigmoid(sn)) = softplus(-sp) + softplus(sn)
    float z0 = -sp, z1 = sn;
    float a0 = (z0 > 30.0f) ? z0 : log1pf(__expf(z0));
    float a1 = (z1 > 30.0f) ? z1 : log1pf(__expf(z1));
    float mk = (float)mask[(size_t)b * LL + (LL - 1)];
    ws[b]        = (a0 + a1) * mk;
    ws[BB + b]   = mk;
  }
}

__global__ __launch_bounds__(256) void din_reduce(const float* __restrict__ ws,
                                                  float* __restrict__ out) {
  __shared__ float sa[256], sm2[256];
  const int t = threadIdx.x;
  float a = 0.0f, m = 0.0f;
  for (int i = t; i < BB; i += 256) {
    a += ws[i];
    m += ws[BB + i];
  }
  sa[t] = a;
  sm2[t] = m;
  __syncthreads();
  for (int s = 128; s > 0; s >>= 1) {
    if (t < s) { sa[t] += sa[t + s]; sm2[t] += sm2[t + s]; }
    __syncthreads();
  }
  if (t == 0) out[BB] = sa[0] / sm2[0];
}

extern "C" void kernel_launch(void* const* d_in, const int* in_sizes, int n_in,
                              void* d_out, int out_size, void* d_ws, size_t ws_size,
                              hipStream_t stream) {
  (void)in_sizes; (void)n_in; (void)out_size; (void)ws_size;
  const float* target = (const float*)d_in[0];
  const float* seq    = (const float*)d_in[1];
  const float* ctx    = (const float*)d_in[2];
  const float* seq2   = (const float*)d_in[3];
  const float* neg    = (const float*)d_in[4];
  const int*   mask   = (const int*)  d_in[5];
  const float* pos    = (const float*)d_in[6];
  const float* wq     = (const float*)d_in[7];
  const float* bq     = (const float*)d_in[8];
  const float* aq     = (const float*)d_in[9];
  const float* w1     = (const float*)d_in[10];
  const float* b1     = (const float*)d_in[11];
  const float* w2     = (const float*)d_in[12];
  const float* b2     = (const float*)d_in[13];
  const float* w3     = (const float*)d_in[14];
  const float* b3     = (const float*)d_in[15];
  const float* wo     = (const float*)d_in[16];
  const float* bo     = (const float*)d_in[17];
  const float* ao     = (const float*)d_in[18];
  float* out = (float*)d_out;
  float* ws  = (float*)d_ws;

  (void)hipFuncSetAttribute(reinterpret_cast<const void*>(din_main),
                            hipFuncAttributeMaxDynamicSharedMemorySize, SMEM_BYTES);
  din_main<<<BB, 256, SMEM_BYTES, stream>>>(target, seq, ctx, seq2, neg, mask, pos,
                                            wq, bq, aq, w1, b1, w2, b2, w3, b3,
                                            wo, bo, ao, out, ws);
  din_reduce<<<1, 256, 0, stream>>>(ws, out);
}